// MVFModuleChannelsLast_33981781246602
// MI455X (gfx1250) — compile-verified
//
#include <hip/hip_runtime.h>

typedef __attribute__((ext_vector_type(2))) float v2f;
typedef __attribute__((ext_vector_type(8))) float v8f;

#define Bn 8
#define Tn 40
#define Hn 32
#define Wn 32
#define Cn 64
// C_MVF = 32, split into 2 WMMA N-tiles of 16; K = 32 split into 8 steps of 4.

__global__ __launch_bounds__(256) void MVFModuleChannelsLast_kernel(
    const float* __restrict__ x,
    const float* __restrict__ wT,   // (3,1,1,32,32) -> [dt][ci][co]
    const float* __restrict__ wH,   // (1,3,1,32,32) -> [dh][ci][co]
    const float* __restrict__ wW,   // (1,1,3,32,32) -> [dw][ci][co]
    float* __restrict__ out)
{
    // ---- stage 9 weight matrices (36 KB) into LDS, pre-swizzled for B frags ----
    // layout index: (((s*8 + kk)*2 + khalf)*32 + n)*2 + j  holds  w_s[ci = kk*4+2*khalf+j][co = n]
    __shared__ float smemB[9 * 1024];
    for (int i = threadIdx.x; i < 9 * 1024; i += 256) {
        int j  = i & 1;
        int n  = (i >> 1) & 31;
        int kh = (i >> 6) & 1;
        int kk = (i >> 7) & 7;
        int s  = i >> 10;
        int k  = kk * 4 + 2 * kh + j;
        const float* wsrc = (s < 3) ? (wT + s * 1024)
                          : (s < 6) ? (wH + (s - 3) * 1024)
                                    : (wW + (s - 6) * 1024);
        smemB[i] = wsrc[k * 32 + n];
    }
    __syncthreads();

    const int lane = threadIdx.x & 31;
    const int m    = lane & 15;   // row within tile / N column
    const int hh   = lane >> 4;   // K-half selector

    // one wave == one tile of 16 consecutive w positions
    int tile = blockIdx.x * 8 + (threadIdx.x >> 5);
    int wt   = tile & 1;
    int tmp  = tile >> 1;
    int h    = tmp & 31; tmp >>= 5;
    int t    = tmp % Tn;
    int b    = tmp / Tn;
    int w0   = wt * 16;

    v8f acc0 = {};   // output channels 0..15
    v8f acc1 = {};   // output channels 16..31

    #pragma unroll
    for (int s = 0; s < 9; ++s) {
        int tt = t, hc = h, dw = 0;
        if (s < 3)      tt = t + (s - 1);
        else if (s < 6) hc = h + (s - 4);
        else            dw = s - 7;
        // wave-uniform at runtime -> EXEC stays all-ones across the WMMAs
        if ((unsigned)tt >= (unsigned)Tn || (unsigned)hc >= (unsigned)Hn) continue;

        int  wl     = w0 + m + dw;                       // per-lane shifted w
        bool lvalid = (unsigned)wl < (unsigned)Wn;       // zero-pad lanes, no divergence at WMMA
        const float* xrow  = x + ((((b * Tn + tt) * Hn + hc) * Wn + wl) * Cn);
        const float* bbase = smemB + (s << 10);

        #pragma unroll
        for (int kk = 0; kk < 8; ++kk) {
            v2f a = {};
            if (lvalid) a = *reinterpret_cast<const v2f*>(xrow + kk * 4 + 2 * hh);
            int bidx = ((kk * 2 + hh) * 32 + m) * 2;
            v2f b0 = *reinterpret_cast<const v2f*>(bbase + bidx);        // co 0..15
            v2f b1 = *reinterpret_cast<const v2f*>(bbase + bidx + 32);   // co 16..31
            acc0 = __builtin_amdgcn_wmma_f32_16x16x4_f32(false, a, false, b0,
                                                         (short)0, acc0, false, false);
            acc1 = __builtin_amdgcn_wmma_f32_16x16x4_f32(false, a, false, b1,
                                                         (short)0, acc1, false, false);
        }
    }

    // ---- ReLU + store conv half, copy skip half ----
    const int    obase = (((b * Tn + t) * Hn + h) * Wn + w0) * Cn;
    float*       orow  = out + obase;
    const float* xob   = x + obase;
    #pragma unroll
    for (int v = 0; v < 8; ++v) {
        int    row = v + 8 * hh;
        float* p   = orow + row * Cn;
        p[m]       = fmaxf(acc0[v], 0.0f);
        p[m + 16]  = fmaxf(acc1[v], 0.0f);
        v2f sk = *reinterpret_cast<const v2f*>(xob + row * Cn + 32 + 2 * m);
        *reinterpret_cast<v2f*>(p + 32 + 2 * m) = sk;
    }
}

extern "C" void kernel_launch(void* const* d_in, const int* in_sizes, int n_in,
                              void* d_out, int out_size, void* d_ws, size_t ws_size,
                              hipStream_t stream) {
    const float* x  = (const float*)d_in[0];
    const float* wT = (const float*)d_in[1];
    const float* wH = (const float*)d_in[2];
    const float* wW = (const float*)d_in[3];
    float* out = (float*)d_out;

    // tiles = B*T*H*(W/16) = 8*40*32*2 = 20480 ; 8 waves (256 thr) per block
    dim3 grid(20480 / 8);
    dim3 block(256);
    MVFModuleChannelsLast_kernel<<<grid, block, 0, stream>>>(x, wT, wH, wW, out);
}